// QLSTM_65481071402959
// MI455X (gfx1250) — compile-verified
//
#include <hip/hip_runtime.h>
#include <math.h>

typedef __attribute__((ext_vector_type(2))) float v2f;
typedef __attribute__((ext_vector_type(4))) float v4f;
typedef __attribute__((ext_vector_type(8))) float v8f;

#define T_DIM 512
#define B_DIM 256
#define D_DIM 512
#define H_DIM 512
#define NQ    8
#define DH    1024

// workspace layout (float offsets)
#define WS_V    0                               // 4*512 folded gate weights
#define WS_MISC 2048                            // beta[4], s[4], cos(th0)[4]
#define WS_A    2560                            // [T*B, 4] gate preactivations
#define WS_H    (2560 + T_DIM * B_DIM * 4)      // [T, B] hidden scalars
#define WS_C    (WS_H + T_DIM * B_DIM)          // [B] final cell scalars

// ---------------------------------------------------------------------------
// Kernel 0: fold weights.  v_g[d] = sum_n Wg0[n]*W_itq[n,d];
// beta_g = Wg0 . b_itq;  s_g = Wg0 . Wh_sum;  cth_g = cos(th_g[0]).
// ---------------------------------------------------------------------------
__global__ void qlstm_fold(const float* __restrict__ W_itq,
                           const float* __restrict__ b_itq,
                           const float* __restrict__ Wg0, const float* __restrict__ th0,
                           const float* __restrict__ Wg1, const float* __restrict__ th1,
                           const float* __restrict__ Wg2, const float* __restrict__ th2,
                           const float* __restrict__ Wg3, const float* __restrict__ th3,
                           float* __restrict__ ws_v, float* __restrict__ ws_misc) {
    __shared__ float wg0[4][NQ];
    __shared__ float whs[NQ];
    const int tid = threadIdx.x;   // 512 threads, 1 block
    if (tid < NQ) {
        wg0[0][tid] = Wg0[tid];    // row 0 of each gate weight matrix
        wg0[1][tid] = Wg1[tid];
        wg0[2][tid] = Wg2[tid];
        wg0[3][tid] = Wg3[tid];
        float s = 0.f;
        const float* p = W_itq + tid * DH + D_DIM;  // hidden-half of row n
        for (int h = 0; h < H_DIM; ++h) s += p[h];
        whs[tid] = s;
    }
    __syncthreads();

    const int d = tid;
    float w[NQ];
#pragma unroll
    for (int n = 0; n < NQ; ++n) w[n] = W_itq[n * DH + d];
#pragma unroll
    for (int g = 0; g < 4; ++g) {
        float acc = 0.f;
#pragma unroll
        for (int n = 0; n < NQ; ++n) acc += wg0[g][n] * w[n];
        ws_v[g * D_DIM + d] = acc;
    }

    if (tid < 4) {
        float beta = 0.f, s = 0.f;
#pragma unroll
        for (int n = 0; n < NQ; ++n) {
            beta += wg0[tid][n] * b_itq[n];
            s    += wg0[tid][n] * whs[n];
        }
        ws_misc[tid]     = beta;
        ws_misc[4 + tid] = s;
        const float* th = (tid == 0) ? th0 : (tid == 1) ? th1 : (tid == 2) ? th2 : th3;
        ws_misc[8 + tid] = cosf(th[0]);
    }
}

// ---------------------------------------------------------------------------
// Kernel 1: projection GEMM via V_WMMA_F32_16X16X4_F32.
// a[row, g] = inputs[row, :] . v_g   (row = t*B + b), N padded 4 -> 16.
// Per-wave 16x16 tile: A 16x4 f32 (lane&15 = M, lane>>4 selects K pair),
// B 4x16 f32 mirrored, 128 WMMA steps over K=512.
// ---------------------------------------------------------------------------
__global__ void __launch_bounds__(256)
qlstm_proj(const float* __restrict__ in,
           const float* __restrict__ ws_v,
           float* __restrict__ ws_a) {
    __shared__ float ldsB[16 * D_DIM];   // 32 KB, rows 4..15 zero-padded
    const int tid = threadIdx.x;
    for (int i = tid; i < 16 * D_DIM; i += 256)
        ldsB[i] = (i < 4 * D_DIM) ? ws_v[i] : 0.0f;
    __syncthreads();

    const int lane    = tid & 31;
    const int wave    = tid >> 5;
    const int n       = lane & 15;   // M for A-frag, N for B/D-frag
    const int half    = lane >> 4;   // selects K pair {0,1} vs {2,3}
    const int rowBase = blockIdx.x * 128 + wave * 16;

    const float* arow = in   + (size_t)(rowBase + n) * D_DIM;
    const float* brow = ldsB + n * D_DIM;

    v8f acc = {};
#pragma unroll 4
    for (int k0 = 0; k0 < D_DIM; k0 += 4) {
        v2f a = *(const v2f*)(arow + k0 + 2 * half);
        v2f b = *(const v2f*)(brow + k0 + 2 * half);
        acc = __builtin_amdgcn_wmma_f32_16x16x4_f32(
            /*neg_a=*/false, a, /*neg_b=*/false, b,
            /*c_mod=*/(short)0, acc, /*reuse_a=*/false, /*reuse_b=*/false);
    }

    // D layout: VGPR r -> lanes 0-15: D[M=r, N=lane]; lanes 16-31: D[M=r+8, N=lane-16]
    if (n < 4) {
#pragma unroll
        for (int r = 0; r < 8; ++r) {
            const int M = r + half * 8;
            ws_a[(size_t)(rowBase + M) * 4 + n] = acc[r];
        }
    }
}

// ---------------------------------------------------------------------------
// Kernel 2: sequential recurrence, one lane per batch element.
// x0_g = a_g + beta_g + h*s_g ; gate = sigmoid(cos(x0_g)*cos(th0_g))
// c = f*c + i*g ; h = o*tanh(c)  (both constant across H).
// ---------------------------------------------------------------------------
__global__ void qlstm_recur(const float* __restrict__ ws_a,
                            const float* __restrict__ ws_misc,
                            float* __restrict__ ws_h,
                            float* __restrict__ ws_c) {
    const int b = blockIdx.x * blockDim.x + threadIdx.x;
    if (b >= B_DIM) return;
    float be[4], s[4], ct[4];
#pragma unroll
    for (int g = 0; g < 4; ++g) {
        be[g] = ws_misc[g];
        s[g]  = ws_misc[4 + g];
        ct[g] = ws_misc[8 + g];
    }
    float h = 0.f, c = 0.f;
    for (int t = 0; t < T_DIM; ++t) {
        v4f a = *(const v4f*)(ws_a + ((size_t)t * B_DIM + b) * 4);
        const float x0 = a.x + be[0] + h * s[0];
        const float x1 = a.y + be[1] + h * s[1];
        const float x2 = a.z + be[2] + h * s[2];
        const float x3 = a.w + be[3] + h * s[3];
        const float gi = 1.f / (1.f + expf(-(cosf(x0) * ct[0])));
        const float gf = 1.f / (1.f + expf(-(cosf(x1) * ct[1])));
        const float gg = 1.f / (1.f + expf(-(cosf(x2) * ct[2])));
        const float go = 1.f / (1.f + expf(-(cosf(x3) * ct[3])));
        c = gf * c + gi * gg;
        h = go * tanhf(c);
        ws_h[(size_t)t * B_DIM + b] = h;
    }
    ws_c[b] = c;
}

// ---------------------------------------------------------------------------
// Kernel 3: broadcast h[t,b] over H into ys, then final hx and cx rows.
// Output layout: ys [T,B,H] ++ hx [B,H] ++ cx [B,H]  (contiguous rows).
// Pure streaming-store pass -> nontemporal float4 stores.
// ---------------------------------------------------------------------------
__global__ void __launch_bounds__(256)
qlstm_bcast(const float* __restrict__ ws_h,
            const float* __restrict__ ws_c,
            float* __restrict__ out) {
    const long gid  = (long)blockIdx.x * blockDim.x + threadIdx.x;
    const long TB   = (long)T_DIM * B_DIM;
    const long ROWS = TB + 2 * B_DIM;
    const long r    = gid >> 7;            // H/4 = 128 lanes per row
    const int  col  = (int)(gid & 127);
    if (r >= ROWS) return;
    float v;
    if (r < TB)                v = ws_h[r];
    else if (r < TB + B_DIM)   v = ws_h[(long)(T_DIM - 1) * B_DIM + (r - TB)];
    else                       v = ws_c[r - TB - B_DIM];
    v4f val = {v, v, v, v};
    __builtin_nontemporal_store(val, (v4f*)out + r * (H_DIM / 4) + col);
}

// ---------------------------------------------------------------------------
extern "C" void kernel_launch(void* const* d_in, const int* in_sizes, int n_in,
                              void* d_out, int out_size, void* d_ws, size_t ws_size,
                              hipStream_t stream) {
    const float* in     = (const float*)d_in[0];
    const float* W_itq  = (const float*)d_in[1];
    const float* b_itq  = (const float*)d_in[2];
    const float* W_in   = (const float*)d_in[3];
    const float* th_in  = (const float*)d_in[4];
    const float* W_fg   = (const float*)d_in[5];
    const float* th_fg  = (const float*)d_in[6];
    const float* W_up   = (const float*)d_in[7];
    const float* th_up  = (const float*)d_in[8];
    const float* W_out  = (const float*)d_in[9];
    const float* th_out = (const float*)d_in[10];
    float* ws  = (float*)d_ws;
    float* out = (float*)d_out;

    qlstm_fold<<<1, 512, 0, stream>>>(W_itq, b_itq,
                                      W_in, th_in, W_fg, th_fg,
                                      W_up, th_up, W_out, th_out,
                                      ws + WS_V, ws + WS_MISC);

    qlstm_proj<<<(T_DIM * B_DIM) / 128, 256, 0, stream>>>(in, ws + WS_V, ws + WS_A);

    qlstm_recur<<<8, 32, 0, stream>>>(ws + WS_A, ws + WS_MISC, ws + WS_H, ws + WS_C);

    const long total  = ((long)T_DIM * B_DIM + 2 * B_DIM) * (H_DIM / 4);
    const int  blocks = (int)((total + 255) / 256);
    qlstm_bcast<<<blocks, 256, 0, stream>>>(ws + WS_H, ws + WS_C, out);
}